// BCNLayer_13108240188142
// MI455X (gfx1250) — compile-verified
//
#include <hip/hip_runtime.h>
#include <cstdint>

#define WIDTH   256
#define HW      (WIDTH * WIDTH)
#define BATCH   256
#define RBAND   4                       // output rows per block band
#define BANDS   (WIDTH / RBAND)         // 64
#define BGROUPS 16                      // batch groups
#define BPB     (BATCH / BGROUPS)       // 16 batches per block
#define TROWS   (RBAND + 2)             // halo rows staged in LDS
#define TILE_FLOATS (TROWS * WIDTH)     // 1536
#define TILE_BYTES  (TILE_FLOATS * 4)   // 6144
#define CHUNKS_PER_ROW (WIDTH * 4 / 8)  // 128 x 8-byte chunks
// 768 chunks total / 256 threads = 3 async b64 issues per thread per batch

typedef int v2i __attribute__((__vector_size__(2 * sizeof(int))));
typedef __attribute__((address_space(1))) v2i gv2i;
typedef __attribute__((address_space(3))) v2i lv2i;

// ---- CDNA5 async global->LDS copy (ASYNCcnt path) ----------------------
__device__ __forceinline__ void async_copy_b64(const void* gsrc, void* lds_dst) {
#if __has_builtin(__builtin_amdgcn_global_load_async_to_lds_b64)
    __builtin_amdgcn_global_load_async_to_lds_b64(
        (gv2i*)(void*)gsrc,
        (lv2i*)lds_dst,
        0, 0);
#else
    unsigned lds_off =
        (unsigned)(uintptr_t)(__attribute__((address_space(3))) void*)lds_dst;
    asm volatile("global_load_async_to_lds_b64 %0, %1, off"
                 :: "v"(lds_off), "v"(gsrc) : "memory");
#endif
}

__device__ __forceinline__ void wait_async3() {
#if __has_builtin(__builtin_amdgcn_s_wait_asynccnt)
    __builtin_amdgcn_s_wait_asynccnt(3);
#else
    asm volatile("s_wait_asynccnt 0x3" ::: "memory");
#endif
}
__device__ __forceinline__ void wait_async0() {
#if __has_builtin(__builtin_amdgcn_s_wait_asynccnt)
    __builtin_amdgcn_s_wait_asynccnt(0);
#else
    asm volatile("s_wait_asynccnt 0x0" ::: "memory");
#endif
}

// -----------------------------------------------------------------------
__global__ __launch_bounds__(256, 1) void bcn_stencil_kernel(
    const float* __restrict__ x,            // (B, HW)
    const float* __restrict__ weights,      // (9, HW)
    const unsigned char* __restrict__ fault_mask, // (HW,) bool
    float* __restrict__ out)                // (B, HW)
{
    __shared__ float tile[2][TILE_FLOATS];  // double-buffered halo tile

    const int j    = threadIdx.x;           // column 0..255
    const int band = blockIdx.x;            // 0..63
    const int r0   = band * RBAND;
    const int b0   = blockIdx.y * BPB;

    // ---- async stage of one batch's halo tile (rows r0-1 .. r0+RBAND) ----
    auto issue = [&](int b, int buf) {
        #pragma unroll
        for (int k = 0; k < 3; ++k) {
            const int c   = j + k * 256;          // 8-byte chunk id, 0..767
            const int tr  = c >> 7;               // tile row 0..5
            const int off = (c & (CHUNKS_PER_ROW - 1)) * 8;
            int grow = r0 - 1 + tr;               // clamp like reference's clip
            grow = grow < 0 ? 0 : (grow > WIDTH - 1 ? WIDTH - 1 : grow);
            const char* gsrc = (const char*)x +
                ((size_t)b * HW + (size_t)grow * WIDTH) * sizeof(float) + off;
            void* ldst = (char*)&tile[buf][0] + (size_t)c * 8;
            async_copy_b64(gsrc, ldst);
        }
    };

    // Kick off the first tile immediately; the fence keeps the async issues
    // ahead of the prologue loads so the longest-latency copy starts first.
    issue(b0, 0);
    asm volatile("" ::: "memory");              // scheduling fence (no insn)

    // ---- preload per-unit weights into registers, zeroing OOB arms ----
    // Weight addresses are always in-bounds; only the arm validity differs,
    // so load unconditionally and mask with a branchless select.
    float wreg[RBAND][9];
    bool  fok[RBAND];
    #pragma unroll
    for (int i = 0; i < RBAND; ++i) {
        const int row = r0 + i;
        const int h   = row * WIDTH + j;
        fok[i] = fault_mask[h] != 0;
        #pragma unroll
        for (int o = 0; o < 9; ++o) {
            const int dy = o / 3 - 1;
            const int dx = o % 3 - 1;
            const bool valid = ((unsigned)(row + dy) < (unsigned)WIDTH) &&
                               ((unsigned)(j   + dx) < (unsigned)WIDTH);
            const float wv = weights[(size_t)o * HW + h];  // unconditional load
            wreg[i][o] = valid ? wv : 0.0f;                // v_cndmask, no branch
        }
    }

    for (int bi = 0; bi < BPB; ++bi) {
        const int cur = bi & 1;
        if (bi + 1 < BPB) {
            issue(b0 + bi + 1, cur ^ 1);          // prefetch next batch
            wait_async3();                        // in-order: current tile done
        } else {
            wait_async0();
        }
        __syncthreads();                          // tile visible to all waves

        const float* t = &tile[cur][0];
        const int jm = (j == 0)         ? 0         : j - 1;
        const int jp = (j == WIDTH - 1) ? WIDTH - 1 : j + 1;

        float xv[TROWS][3];
        #pragma unroll
        for (int tr = 0; tr < TROWS; ++tr) {
            xv[tr][0] = t[tr * WIDTH + jm];
            xv[tr][1] = t[tr * WIDTH + j];
            xv[tr][2] = t[tr * WIDTH + jp];
        }

        const int b = b0 + bi;
        #pragma unroll
        for (int i = 0; i < RBAND; ++i) {
            float acc = 0.0f;
            #pragma unroll
            for (int o = 0; o < 9; ++o)
                acc = fmaf(wreg[i][o], xv[i + o / 3][o % 3], acc);
            if (!fok[i]) acc = 0.0f;              // fault -> sigmoid(0)=0.5
            const float s = 1.0f / (1.0f + __expf(-acc));
            __builtin_nontemporal_store(
                s, &out[(size_t)b * HW + (size_t)(r0 + i) * WIDTH + j]);
        }
        __syncthreads();                          // protect buffer reuse
    }
}

// -----------------------------------------------------------------------
extern "C" void kernel_launch(void* const* d_in, const int* in_sizes, int n_in,
                              void* d_out, int out_size, void* d_ws, size_t ws_size,
                              hipStream_t stream) {
    // setup_inputs order: x, weights, nbr_idx, valid, fault_mask
    const float*         x          = (const float*)d_in[0];
    const float*         weights    = (const float*)d_in[1];
    // nbr_idx (d_in[2]) and valid (d_in[3]) are analytic -> unused
    const unsigned char* fault_mask = (const unsigned char*)d_in[4];
    float*               out        = (float*)d_out;

    dim3 grid(BANDS, BGROUPS);
    dim3 block(256);
    bcn_stencil_kernel<<<grid, block, 0, stream>>>(x, weights, fault_mask, out);
}